// MultiDomainInformationFusion_59124519797139
// MI455X (gfx1250) — compile-verified
//
#include <hip/hip_runtime.h>
#include <hip/hip_bf16.h>

typedef __attribute__((ext_vector_type(2))) float v2f;
typedef __attribute__((ext_vector_type(8))) float v8f;

// ---------------- degree / norm ----------------
__global__ void deg_init_kernel(float* deg, int n) {
    int i = blockIdx.x * blockDim.x + threadIdx.x;
    if (i < n) deg[i] = 1.0f;  // self-loop
}

__global__ void deg_count_kernel(const int* __restrict__ dst, float* deg, int E) {
    int e = blockIdx.x * blockDim.x + threadIdx.x;
    if (e < E) atomicAdd(&deg[dst[e]], 1.0f);
}

__global__ void dinv_kernel(float* deg, int n) {
    int i = blockIdx.x * blockDim.x + threadIdx.x;
    if (i < n) {
        float d = deg[i];
        deg[i] = (d > 0.0f) ? rsqrtf(d) : 0.0f;  // in place: deg -> dinv
    }
}

// ---------------- zero ----------------
__global__ void zero_kernel(float* p, long n) {
    long i = (long)blockIdx.x * blockDim.x + threadIdx.x;
    if (i < n) p[i] = 0.0f;
}

// ---------------- fp32 WMMA GEMM: H[M,NOUT] = X[M,128] @ W[128,NOUT] ----------------
// One wave computes a 16-row tile across the full NOUT width.
// V_WMMA_F32_16X16X4_F32 fragment layout (ISA 7.12.2):
//   A (16x4):  lane%16 = M row; lanes 0-15 hold K={0,1} in v[0],v[1]; lanes 16-31 hold K={2,3}
//   B (4x16):  lane%16 = N col; same K split as A
//   C/D 16x16: lane%16 = N col; VGPR j holds row M = j + 8*(lane/16)
template <int NOUT>
__global__ void gemm_wmma_kernel(const float* __restrict__ X,
                                 const float* __restrict__ W,
                                 float* __restrict__ H, int M) {
    constexpr int NT = NOUT / 16;
    const int wave = threadIdx.x >> 5;
    const int lane = threadIdx.x & 31;
    const int tileM = blockIdx.x * (blockDim.x >> 5) + wave;
    const int numTiles = (M + 15) >> 4;
    if (tileM >= numTiles) return;  // uniform per wave

    const int m15 = lane & 15;
    const int half = lane >> 4;
    int row = tileM * 16 + m15;
    if (row >= M) row = M - 1;  // safe (M is a multiple of 16 here anyway)

    const v8f vzero = {0.f, 0.f, 0.f, 0.f, 0.f, 0.f, 0.f, 0.f};
    v8f acc[NT];
#pragma unroll
    for (int t = 0; t < NT; ++t) acc[t] = vzero;

    const float* xrow = X + (size_t)row * 128;
#pragma unroll 4
    for (int k0 = 0; k0 < 128; k0 += 4) {
        const int kk = k0 + 2 * half;
        v2f a = {xrow[kk], xrow[kk + 1]};
#pragma unroll
        for (int t = 0; t < NT; ++t) {
            const int col = t * 16 + m15;
            v2f b = {W[(size_t)kk * NOUT + col], W[(size_t)(kk + 1) * NOUT + col]};
            acc[t] = __builtin_amdgcn_wmma_f32_16x16x4_f32(
                false, a, false, b, (short)0, acc[t], false, false);
        }
    }

#pragma unroll
    for (int t = 0; t < NT; ++t) {
        const int col = t * 16 + m15;
#pragma unroll
        for (int j = 0; j < 8; ++j) {
            const int r = tileM * 16 + j + 8 * half;
            if (r < M) H[(size_t)r * NOUT + col] = acc[t][j];
        }
    }
}

// ---------------- edge scatter: agg[dst] += h[src] * dinv[src]*dinv[dst] ----------------
template <int D>
__global__ void scatter_kernel(const float* __restrict__ h,
                               const int* __restrict__ src,
                               const int* __restrict__ dst,
                               const float* __restrict__ dinv,
                               float* __restrict__ agg, int E) {
    constexpr int CH = D / 4;  // float4 chunks per edge
    long idx = (long)blockIdx.x * blockDim.x + threadIdx.x;
    if (idx >= (long)E * CH) return;
    const int e = (int)(idx / CH);
    const int c = ((int)(idx % CH)) * 4;
    const int s = src[e];
    const int d = dst[e];
    const float nrm = dinv[s] * dinv[d];
    const float4 v = *(const float4*)(h + (size_t)s * D + c);
    float* o = agg + (size_t)d * D + c;
    atomicAdd(o + 0, v.x * nrm);
    atomicAdd(o + 1, v.y * nrm);
    atomicAdd(o + 2, v.z * nrm);
    atomicAdd(o + 3, v.w * nrm);
}

// ---------------- finalize: out = agg + h*dinv^2 + bias (, relu) ----------------
template <int D, bool RELU>
__global__ void finalize_kernel(float* __restrict__ agg,
                                const float* __restrict__ h,
                                const float* __restrict__ dinv,
                                const float* __restrict__ bias, int n) {
    long i = (long)blockIdx.x * blockDim.x + threadIdx.x;
    if (i >= (long)n * D) return;
    const int node = (int)(i / D);
    const int f = (int)(i % D);
    const float s = dinv[node];
    float v = agg[i] + h[i] * s * s + bias[f];
    if (RELU) v = fmaxf(v, 0.0f);
    agg[i] = v;
}

static inline int cdiv(long a, int b) { return (int)((a + b - 1) / b); }

extern "C" void kernel_launch(void* const* d_in, const int* in_sizes, int n_in,
                              void* d_out, int out_size, void* d_ws, size_t ws_size,
                              hipStream_t stream) {
    const float* x  = (const float*)d_in[0];
    const int*   ei = (const int*)d_in[1];
    const float* W1 = (const float*)d_in[2];
    const float* b1 = (const float*)d_in[3];
    const float* W2 = (const float*)d_in[4];
    const float* b2 = (const float*)d_in[5];

    const int N = in_sizes[0] / 128;  // 50000
    const int E = in_sizes[1] / 2;    // 500000
    const int* src = ei;
    const int* dst = ei + E;

    // workspace layout (floats)
    float* ws   = (float*)d_ws;
    float* dinv = ws;                         // N
    float* h1   = dinv + N;                   // N*128
    float* agg1 = h1 + (size_t)N * 128;       // N*128 (becomes relu'd layer-1 output)
    float* h2   = agg1 + (size_t)N * 128;     // N*64
    float* out  = (float*)d_out;              // N*64, used as agg2 then final

    const int B = 256;

    // degrees -> dinv (in place)
    deg_init_kernel<<<cdiv(N, B), B, 0, stream>>>(dinv, N);
    deg_count_kernel<<<cdiv(E, B), B, 0, stream>>>(dst, dinv, E);
    dinv_kernel<<<cdiv(N, B), B, 0, stream>>>(dinv, N);

    // ---- layer 1 ----
    const int tiles = (N + 15) / 16;
    const int wavesPerBlock = B / 32;
    gemm_wmma_kernel<128><<<cdiv(tiles, wavesPerBlock), B, 0, stream>>>(x, W1, h1, N);
    zero_kernel<<<cdiv((long)N * 128, B), B, 0, stream>>>(agg1, (long)N * 128);
    scatter_kernel<128><<<cdiv((long)E * 32, B), B, 0, stream>>>(h1, src, dst, dinv, agg1, E);
    finalize_kernel<128, true><<<cdiv((long)N * 128, B), B, 0, stream>>>(agg1, h1, dinv, b1, N);

    // ---- layer 2 ----
    gemm_wmma_kernel<64><<<cdiv(tiles, wavesPerBlock), B, 0, stream>>>(agg1, W2, h2, N);
    zero_kernel<<<cdiv((long)N * 64, B), B, 0, stream>>>(out, (long)N * 64);
    scatter_kernel<64><<<cdiv((long)E * 16, B), B, 0, stream>>>(h2, src, dst, dinv, out, E);
    finalize_kernel<64, false><<<cdiv((long)N * 64, B), B, 0, stream>>>(out, h2, dinv, b2, N);
}